// MomentContraction_44830868635785
// MI455X (gfx1250) — compile-verified
//
#include <hip/hip_runtime.h>
#include <stdint.h>

// ---------------------------------------------------------------------------
// MomentContraction for MI455X (gfx1250, wave32)
//   out[(p*8+t)*A + a] = sum_{i,j} m1[a,r,i]*m1[a,s,j]*m2[a,t,i,j],
//   p = tril pair index of (r,s), r >= s, p = r(r+1)/2 + s.
// Per atom: P(36x9) x Q(9x8) on V_WMMA_F32_16X16X4_F32 (9 WMMAs/atom).
// Inputs staged global->LDS with async b128 copies (ASYNCcnt, double
// buffered); outputs staged through LDS into feature-major, atom-contiguous
// coalesced b128 stores.  All lane predication done with address/value
// selects (no EXEC manipulation in the hot loop).
// ---------------------------------------------------------------------------

#define N_ATOMS   250000
#define N_TRIL    36
#define TILE      8                      // atoms per wave iteration
#define WPB       4                      // waves per block
#define BLOCK     (WPB * 32)
#define N_TILES   (N_ATOMS / TILE)       // 31250, exact
#define IN_FLOATS   (TILE * 96)          // 768  (24 m1 + 72 m2 per atom)
#define OUT_FLOATS  (TILE * 288)         // 2304
#define WAVE_LDS    (2 * IN_FLOATS + OUT_FLOATS)  // 3840 floats
#define DUMP_FLOATS 512                  // absorbs lane + v*64 redirects

typedef __attribute__((ext_vector_type(2))) float v2f;
typedef __attribute__((ext_vector_type(4))) float v4f;
typedef __attribute__((ext_vector_type(8))) float v8f;
typedef __attribute__((ext_vector_type(4))) int   v4i;

// ---- CDNA5 async global->LDS copy (ASYNCcnt) with safe fallback -----------
#if defined(__gfx1250__) && __has_builtin(__builtin_amdgcn_global_load_async_to_lds_b128) && __has_builtin(__builtin_amdgcn_s_wait_asynccnt)
#define HAVE_ASYNC 1
typedef __attribute__((address_space(3))) v4i* lds_v4i_t;
#define ASYNC_CP16(dst, src)                                                   \
  __builtin_amdgcn_global_load_async_to_lds_b128(                              \
      (v4i*)(uintptr_t)(const void*)(src),                                     \
      (lds_v4i_t)(uintptr_t)(void*)(dst), 0, 0)
#define WAIT_ASYNC(n) do { __builtin_amdgcn_s_wait_asynccnt(n); } while (0)
#else
#define HAVE_ASYNC 0
#define ASYNC_CP16(dst, src) do { *(v4f*)(dst) = *(const v4f*)(src); } while (0)
#define WAIT_ASYNC(n) do { } while (0)
#endif

// Issue exactly 7 async b128 instructions per tile (2 for m1, 5 for m2).
// Lanes past the last transfer are clamped to a duplicate (benign) transfer
// so EXEC stays full and the ASYNCcnt issue count is deterministic.
__device__ __forceinline__ void issue_tile_loads(int tile, float* dst,
                                                 const float* m1,
                                                 const float* m2, int lane) {
  const size_t a0 = (size_t)tile * TILE;
  const float* g1 = m1 + a0 * 24;   // 192 floats = 48 x b128
  const float* g2 = m2 + a0 * 72;   // 576 floats = 144 x b128
#pragma unroll
  for (int r = 0; r < 2; ++r) {
    int tr = r * 32 + lane; if (tr > 47) tr = 47;
    ASYNC_CP16(dst + (tr / 6) * 96 + (tr % 6) * 4, g1 + tr * 4);
  }
#pragma unroll
  for (int r = 0; r < 5; ++r) {
    int tr = r * 32 + lane; if (tr > 143) tr = 143;
    ASYNC_CP16(dst + (tr / 18) * 96 + 24 + (tr % 18) * 4, g2 + tr * 4);
  }
}

__global__ __launch_bounds__(BLOCK) void
moment_contraction_wmma(const float* __restrict__ m1,
                        const float* __restrict__ m2,
                        float* __restrict__ out,
                        int n_waves_total) {
  __shared__ float lds[WPB * WAVE_LDS + DUMP_FLOATS];  // 62 KB

  const int tid  = threadIdx.x;
  const int wave = tid >> 5;
  const int lane = tid & 31;
  const int lid  = lane & 15;
  const bool hi  = lane >= 16;

  float* wbase  = &lds[wave * WAVE_LDS];
  float* inBuf0 = wbase;
  float* inBuf1 = wbase + IN_FLOATS;
  float* outB   = wbase + 2 * IN_FLOATS;
  float* dump   = &lds[WPB * WAVE_LDS];   // shared garbage strip

  // ---- per-lane constant prep (hoisted out of the atom loop) ----
  // A-matrix (16x4 f32): lanes 0-15 hold rows 0-15 with K=0,1 in the two
  // VGPRs; lanes 16-31 hold the same rows with K=2,3.  Row = tril pair.
  int  rO[3], sO[3];
  bool rowOK[3];
#pragma unroll
  for (int mt = 0; mt < 3; ++mt) {
    int row = mt * 16 + lid;
    rowOK[mt] = row < N_TRIL;
    int rowc = rowOK[mt] ? row : 0;          // clamp -> safe LDS offsets
    int rr = 0;
    while (((rr + 1) * (rr + 2)) / 2 <= rowc) ++rr;   // <= 7 iters, once
    rO[mt] = rr * 3;
    sO[mt] = (rowc - (rr * (rr + 1)) / 2) * 3;
  }

  // B-matrix (4x16 f32), mirrored layout: lane holds column N = lid,
  // K = kk*4 + (hi?2:0) + {0,1}.  Q[k, t] = m2flat[t*9 + k]; zero-pad
  // K>=9 and t>=8 on the B side (A's padded-K values then don't matter).
  const int  t   = lid;
  const bool tOK = t < 8;
  int  bO0[3], bO1[3];
  bool bZ0[3], bZ1[3];
#pragma unroll
  for (int kk = 0; kk < 3; ++kk) {
    int k0 = kk * 4 + (hi ? 2 : 0);
    bZ0[kk] = tOK && (k0 < 9);
    bZ1[kk] = tOK && (k0 + 1 < 9);
    bO0[kk] = bZ0[kk] ? (24 + t * 9 + k0)     : 0;   // clamped safe offsets
    bO1[kk] = bZ1[kk] ? (24 + t * 9 + k0 + 1) : 0;
  }

  // C scatter bases: feature f = row*8 + t, LDS index f*8 + slot.
  // Invalid lanes redirect to the dump strip (address select, no EXEC ops).
  int  cBase[3];
  bool cOK[3];
#pragma unroll
  for (int mt = 0; mt < 3; ++mt) {
    cBase[mt] = ((mt * 16 + (hi ? 8 : 0)) * 8 + t) * 8;
    cOK[mt]   = (mt == 2) ? (tOK && !hi) : tOK;
  }

  const int gwave = blockIdx.x * WPB + wave;

  // ---- software-pipelined tile loop (double-buffered async prefetch) ----
  int tile = gwave;
  if (tile < N_TILES) issue_tile_loads(tile, inBuf0, m1, m2, lane);

  int buf = 0;
  for (; tile < N_TILES; tile += n_waves_total) {
    float* cur = buf ? inBuf1 : inBuf0;
    float* nbf = buf ? inBuf0 : inBuf1;
    const int nxt = tile + n_waves_total;
    if (nxt < N_TILES) {
      issue_tile_loads(nxt, nbf, m1, m2, lane);
      WAIT_ASYNC(7);   // async loads complete in order: current tile ready
    } else {
      WAIT_ASYNC(0);
    }

    const size_t a0 = (size_t)tile * TILE;

#pragma unroll 1
    for (int slot = 0; slot < TILE; ++slot) {
      const float* ap = cur + slot * 96;  // [0..23]=m1[r][i], [24..95]=m2[t][i][j]

      // B operands: unconditional clamped loads, then value selects.
      v2f B[3];
#pragma unroll
      for (int kk = 0; kk < 3; ++kk) {
        float b0 = ap[bO0[kk]];
        float b1 = ap[bO1[kk]];
        B[kk] = (v2f){bZ0[kk] ? b0 : 0.0f, bZ1[kk] ? b1 : 0.0f};
      }

      v8f C[3] = {};
#pragma unroll
      for (int mt = 0; mt < 3; ++mt) {
        // unconditional loads from clamped row offsets
        float r0 = ap[rO[mt] + 0], r1 = ap[rO[mt] + 1], r2 = ap[rO[mt] + 2];
        float s0 = ap[sO[mt] + 0], s1 = ap[sO[mt] + 1], s2 = ap[sO[mt] + 2];
        // P[row, k=(i*3+j)] = m1[r,i]*m1[s,j]
        float p00 = r0 * s0, p01 = r0 * s1, p02 = r0 * s2;
        float p10 = r1 * s0, p11 = r1 * s1, p12 = r1 * s2;
        float p20 = r2 * s0, p21 = r2 * s1, p22 = r2 * s2;
        // pack A per K-step: lo half carries k=kk*4+{0,1}, hi half k=kk*4+{2,3}
        v2f A0 = hi ? (v2f){p02, p10} : (v2f){p00, p01};
        v2f A1 = hi ? (v2f){p20, p21} : (v2f){p11, p12};
        v2f A2 = hi ? (v2f){0.0f, 0.0f} : (v2f){p22, 0.0f};  // k>=9: B is zero
        if (mt == 2) {                       // zero padded rows 36..47
          const v2f z = (v2f){0.0f, 0.0f};
          const bool ok = rowOK[2];
          A0 = ok ? A0 : z;  A1 = ok ? A1 : z;  A2 = ok ? A2 : z;
        }

        C[mt] = __builtin_amdgcn_wmma_f32_16x16x4_f32(
            false, A0, false, B[0], (short)0, C[mt], false, false);
        C[mt] = __builtin_amdgcn_wmma_f32_16x16x4_f32(
            false, A1, false, B[1], (short)0, C[mt], false, false);
        C[mt] = __builtin_amdgcn_wmma_f32_16x16x4_f32(
            false, A2, false, B[2], (short)0, C[mt], false, false);
      }

      // Scatter C into [feature][slot]; invalid lanes write the dump strip.
      // Plain ds_store_b32 with immediate offsets v*256B, no EXEC changes.
#pragma unroll
      for (int mt = 0; mt < 3; ++mt) {
        float* sp = cOK[mt] ? (outB + cBase[mt] + slot) : (dump + lane);
        const int vmax = (mt == 2) ? 4 : 8;   // rows >= 36 skipped at build
#pragma unroll
        for (int v = 0; v < vmax; ++v) sp[v * 64] = C[mt][v];
      }
    }  // slot

    // Coalesced, atom-contiguous stores: out[f*A + a0 + sl], 16B aligned.
#pragma unroll 2
    for (int c = 0; c < 18; ++c) {
      int idx4 = c * 128 + lane * 4;   // 2304 floats per tile
      int f    = idx4 >> 3;
      int sl   = idx4 & 7;             // 0 or 4
      v4f val  = *(const v4f*)&outB[idx4];
      *(v4f*)(out + (size_t)f * N_ATOMS + a0 + sl) = val;
    }

    buf ^= 1;
  }
}

extern "C" void kernel_launch(void* const* d_in, const int* in_sizes, int n_in,
                              void* d_out, int out_size, void* d_ws,
                              size_t ws_size, hipStream_t stream) {
  (void)in_sizes; (void)n_in; (void)out_size; (void)d_ws; (void)ws_size;
  const float* m1 = (const float*)d_in[0];
  const float* m2 = (const float*)d_in[1];
  float* out = (float*)d_out;

  const int blocks = 1024;                 // 4096 waves, ~7.6 tiles each
  const int waves  = blocks * WPB;
  hipLaunchKernelGGL(moment_contraction_wmma, dim3(blocks), dim3(BLOCK), 0,
                     stream, m1, m2, out, waves);
}